// MatchNetLoss_70016556859475
// MI455X (gfx1250) — compile-verified
//
#include <hip/hip_runtime.h>
#include <hip/hip_bf16.h>

typedef float v2f __attribute__((ext_vector_type(2)));
typedef float v8f __attribute__((ext_vector_type(8)));

#define BS_   32
#define V_    4096
#define NGT_  256
#define THRESH_ 0.01f
#define EPS_  1e-7f

// workspace layout (floats):
//   ws[0..31]   per-sample masked distance sum
//   ws[32..63]  per-sample mask count
//   ws[64]      tp (true-positive count)
//   ws[65]      bce sum
__global__ void mnl_init_ws(float* ws) {
    int i = threadIdx.x;
    if (i < 66) ws[i] = 0.0f;
}

__global__ void mnl_bce(const float* __restrict__ probs,
                        const float* __restrict__ target,
                        float* __restrict__ ws) {
    int i = blockIdx.x * blockDim.x + threadIdx.x;
    float v = 0.0f;
    if (i < BS_ * V_) {
        float p = probs[i];
        p = fminf(fmaxf(p, EPS_), 1.0f - EPS_);
        float t = target[i];
        v = -(t * logf(p) + (1.0f - t) * log1pf(-p));
    }
    // wave32 reduce
    #pragma unroll
    for (int m = 16; m > 0; m >>= 1) v += __shfl_xor(v, m, 32);
    if ((threadIdx.x & 31) == 0) atomicAdd(&ws[65], v);
}

// One wave per (batch, 16-voxel tile). Distance matrix tile 16(M voxels) x 16(N gt)
// computed with V_WMMA_F32_16X16X4_F32 via the norm expansion. We minimize the
// SQUARED distance (sqrt is monotonic) and take one sqrt per voxel at the end.
__global__ __launch_bounds__(256) void mnl_chamfer(
        const float* __restrict__ probs,
        const float* __restrict__ target,
        const float* __restrict__ gt,       // (BS, NGT, 3)
        const float* __restrict__ centers,  // (V, 3)
        float* __restrict__ ws) {
    const int lane = threadIdx.x & 31;
    const int wave = threadIdx.x >> 5;
    const int b    = blockIdx.x >> 5;     // 32 batches
    const int tg   = blockIdx.x & 31;     // 32 tile-groups of 128 voxels
    const int vbase = tg * 128 + wave * 16;
    const int hf = lane >> 4;             // which 16-lane half
    const int ln = lane & 15;

    // --- A matrix: 16x4 f32 (M=voxels, K = x,y,z,0) ---
    // lanes 0-15: VGPR0=K0(x), VGPR1=K1(y); lanes 16-31: VGPR0=K2(z), VGPR1=K3(0)
    const float* cA = centers + (size_t)(vbase + ln) * 3;
    v2f amat;
    amat.x = hf ? cA[2] : cA[0];
    amat.y = hf ? 0.0f  : cA[1];

    // ||c_m||^2 for the 8 rows this lane sees in the D layout (m = r + hf*8)
    float cn2[8];
    #pragma unroll
    for (int r = 0; r < 8; ++r) {
        const float* c = centers + (size_t)(vbase + hf * 8 + r) * 3;
        cn2[r] = c[0] * c[0] + c[1] * c[1] + c[2] * c[2];
    }

    float d2min[8];
    #pragma unroll
    for (int r = 0; r < 8; ++r) d2min[r] = 3.4e38f;

    const float* gtb = gt + (size_t)b * NGT_ * 3;
    for (int nt = 0; nt < NGT_ / 16; ++nt) {
        // --- B matrix: 4x16 f32 (K rows x N gt points) ---
        // lanes 0-15: VGPR0=K0(x), VGPR1=K1(y); lanes 16-31: VGPR0=K2(z), VGPR1=K3(0)
        const float* g = gtb + (size_t)(nt * 16 + ln) * 3;
        float gx = g[0], gy = g[1], gz = g[2];
        v2f bmat;
        bmat.x = hf ? gz   : gx;
        bmat.y = hf ? 0.0f : gy;
        float gn2 = gx * gx + gy * gy + gz * gz;   // per-lane: n = ln (both halves)

        v8f cz = {};
        v8f acc = __builtin_amdgcn_wmma_f32_16x16x4_f32(
            /*neg_a=*/false, amat, /*neg_b=*/false, bmat,
            /*c_mod=*/(short)0, cz, /*reuse_a=*/false, /*reuse_b=*/false);

        #pragma unroll
        for (int r = 0; r < 8; ++r) {
            float d2 = (cn2[r] + gn2) - 2.0f * acc[r];
            d2min[r] = fminf(d2min[r], d2);
        }
    }

    // min over the 16 N-columns: xor offsets 1,2,4,8 stay within each 16-lane half
    #pragma unroll
    for (int r = 0; r < 8; ++r) {
        #pragma unroll
        for (int m = 8; m > 0; m >>= 1)
            d2min[r] = fminf(d2min[r], __shfl_xor(d2min[r], m, 32));
    }

    // lanes with ln<8 each own one voxel: m = hf*8 + ln, value = d2min[ln]
    float d2m = d2min[0];
    #pragma unroll
    for (int r = 1; r < 8; ++r) d2m = (ln == r) ? d2min[r] : d2m;

    float s = 0.0f, cnt = 0.0f, tp = 0.0f;
    if (ln < 8) {
        int vox = vbase + hf * 8 + ln;
        float p = probs[(size_t)b * V_ + vox];
        float t = target[(size_t)b * V_ + vox];
        if (p > THRESH_) {
            s = __builtin_sqrtf(fmaxf(d2m, 0.0f));  // single sqrt per voxel
            cnt = 1.0f;
            if (t > 0.5f) tp = 1.0f;
        }
    }
    #pragma unroll
    for (int m = 16; m > 0; m >>= 1) {
        s   += __shfl_xor(s, m, 32);
        cnt += __shfl_xor(cnt, m, 32);
        tp  += __shfl_xor(tp, m, 32);
    }
    if (lane == 0) {
        atomicAdd(&ws[b], s);
        atomicAdd(&ws[32 + b], cnt);
        atomicAdd(&ws[64], tp);
    }
}

__global__ void mnl_finalize(const float* __restrict__ ws, float* __restrict__ out) {
    if (threadIdx.x == 0 && blockIdx.x == 0) {
        float cd = 0.0f;
        for (int b = 0; b < BS_; ++b) {
            float cnt = ws[32 + b];
            if (cnt > 0.0f) cd += ws[b] / cnt;   // cnt>0 => max(cnt,1)==cnt
        }
        float c_loss = cd / (float)BS_;
        float pred_loss = ws[65] / (float)(BS_ * V_);
        float tp = ws[64];
        out[0] = pred_loss + c_loss + expf(-tp);
    }
}

extern "C" void kernel_launch(void* const* d_in, const int* in_sizes, int n_in,
                              void* d_out, int out_size, void* d_ws, size_t ws_size,
                              hipStream_t stream) {
    const float* probs   = (const float*)d_in[0];  // (32, 4096)
    const float* target  = (const float*)d_in[1];  // (32, 4096)
    const float* diff_gt = (const float*)d_in[2];  // (32, 256, 3)
    const float* centers = (const float*)d_in[3];  // (4096, 3)
    float* ws  = (float*)d_ws;
    float* out = (float*)d_out;

    mnl_init_ws<<<1, 128, 0, stream>>>(ws);
    mnl_bce<<<(BS_ * V_ + 255) / 256, 256, 0, stream>>>(probs, target, ws);
    mnl_chamfer<<<BS_ * 32, 256, 0, stream>>>(probs, target, diff_gt, centers, ws);
    mnl_finalize<<<1, 32, 0, stream>>>(ws, out);
}